// CapsuleLinear_59201829208534
// MI455X (gfx1250) — compile-verified
//
#include <hip/hip_runtime.h>
#include <math.h>

// CapsuleLinear with k-means ('dot') routing for MI455X (gfx1250, wave32).
// B=64 batches, I=512 in-capsules, O=64 out-capsules, L=8 in-dim, V=16 out-dim.
//
// priors (128 MB) are never materialized:
//   init:   out[(o,v),b] = (1/I) * W[1024x4096] @ Xf[4096x64]   (WMMA f32 16x16x4)
//   logits: 16x16 prior tiles (2 WMMAs) dotted with nhat, softmax over O in LDS
//   update: per o: W_o[16x4096] @ (x*probs_o)[4096x64]          (WMMA f32 16x16x4)
// GEMM kernels: one WG per o, 4 waves K-split, each wave accumulates ALL four
// 16-batch tiles against a single A (weight) load -> weight stream read once
// per pass device-wide (16 MB), 4 WMMAs per 5 b64 loads.

typedef float v2f __attribute__((ext_vector_type(2)));
typedef float v4f __attribute__((ext_vector_type(4)));
typedef float v8f __attribute__((ext_vector_type(8)));

#define BATCH 64
#define NI    512
#define NO    64
#define NV    16
#define NL    8
#define KTOT  (NI * NL)   // 4096
#define KSPL  (KTOT / 4)  // 1024 per wave
#define IB    8           // i-block per logits workgroup

static __device__ __forceinline__ v8f wmma_f32_k4(v2f a, v2f b, v8f c) {
  // D = A(16x4) * B(4x16) + C, all f32.
  // (neg_a, A, neg_b, B, c_mod, C, reuse_a, reuse_b)
  return __builtin_amdgcn_wmma_f32_16x16x4_f32(false, a, false, b, (short)0, c,
                                               false, false);
}

static __device__ __forceinline__ void stage_tile(float* red, int slot, int lane,
                                                  v8f c) {
  v4f lo = __builtin_shufflevector(c, c, 0, 1, 2, 3);
  v4f hi = __builtin_shufflevector(c, c, 4, 5, 6, 7);
  *(v4f*)&red[((size_t)slot * 32 + lane) * 8 + 0] = lo;   // ds_store_b128
  *(v4f*)&red[((size_t)slot * 32 + lane) * 8 + 4] = hi;
}

// ---------------------------------------------------------------------------
// GEMM body shared by init/update. One WG per o (4 waves, K-split); each wave
// accumulates 4 batch-tiles reusing one A load; 16 KB LDS reduction.
// scale_probs: multiply B operand by probs[o,i,b] (update) or not (init).
// ---------------------------------------------------------------------------
template <bool WITH_PROBS>
static __device__ __forceinline__ void caps_gemm_body(
    const float* __restrict__ w, const float* __restrict__ x,
    const float* __restrict__ probs, float* __restrict__ outw, float out_scale) {
  __shared__ float red[4 * 4 * 32 * 8];   // [ks][bt][lane][r] = 16 KB
  const int lane = threadIdx.x & 31;
  const int ks   = threadIdx.x >> 5;      // K-split slice 0..3
  const int o    = blockIdx.x;
  const int half = lane >> 4;             // K sub-pair select (ISA A/B layout)
  const int lr   = lane & 15;             // row (A: v) / col (B,D: b)
  const float* __restrict__ wo = w + (size_t)o * NI * NV * NL;   // [i][v][l]
  const float* __restrict__ po =
      WITH_PROBS ? probs + (size_t)o * NI * BATCH : nullptr;     // [i][b]

  v8f c0 = {}, c1 = {}, c2 = {}, c3 = {};
  const int kbeg = ks * KSPL, kend = kbeg + KSPL;
  for (int k0 = kbeg; k0 < kend; k0 += 4) {
    const int k = k0 + 2 * half;
    const int i = k >> 3;
    const int l = k & 7;
    if ((k0 & 255) == 0 && (k0 + 512) < kend)
      __builtin_prefetch(wo + (size_t)((k0 + 512) >> 3) * NV * NL, 0, 0);

    v2f a = *(const v2f*)(wo + ((size_t)i * NV + lr) * NL + l);  // K,K+1 (b64)

    const float* xb = x + ((size_t)lr * NI + i) * NL + l;        // bt=0 col lr
    v2f b0 = *(const v2f*)(xb);
    v2f b1 = *(const v2f*)(xb + (size_t)16 * NI * NL);
    v2f b2 = *(const v2f*)(xb + (size_t)32 * NI * NL);
    v2f b3 = *(const v2f*)(xb + (size_t)48 * NI * NL);
    if (WITH_PROBS) {
      const float* pb = po + (size_t)i * BATCH + lr;
      b0 = b0 * pb[0];
      b1 = b1 * pb[16];
      b2 = b2 * pb[32];
      b3 = b3 * pb[48];
    }
    c0 = wmma_f32_k4(a, b0, c0);   // one A feeds 4 WMMAs (operand reuse)
    c1 = wmma_f32_k4(a, b1, c1);
    c2 = wmma_f32_k4(a, b2, c2);
    c3 = wmma_f32_k4(a, b3, c3);
  }

  stage_tile(red, ks * 4 + 0, lane, c0);
  stage_tile(red, ks * 4 + 1, lane, c1);
  stage_tile(red, ks * 4 + 2, lane, c2);
  stage_tile(red, ks * 4 + 3, lane, c3);
  __syncthreads();

  // wave ks reduces batch-tile bt == ks across the 4 K-slices
  const int bt = ks;
  const int bb = bt * 16 + lr;
  v4f alo = {}, ahi = {};
  #pragma unroll
  for (int s = 0; s < 4; ++s) {
    const float* q = &red[((size_t)(s * 4 + bt) * 32 + lane) * 8];
    alo += *(const v4f*)(q);      // ds_load_b128
    ahi += *(const v4f*)(q + 4);
  }
  #pragma unroll
  for (int r = 0; r < 4; ++r) {
    outw[((size_t)o * NV + (r + 8 * half)) * BATCH + bb]     = alo[r] * out_scale;
    outw[((size_t)o * NV + (r + 4 + 8 * half)) * BATCH + bb] = ahi[r] * out_scale;
  }
}

// out[(o,v),b] = (1/I) * sum_{(i,l)} W[o,i,v,l] * x[b,i,l]
__global__ void caps_init_kernel(const float* __restrict__ w,
                                 const float* __restrict__ x,
                                 float* __restrict__ outw) {
  caps_gemm_body<false>(w, x, nullptr, outw, 1.0f / (float)NI);
}

// out[(o,v),b] = sum_{(i,l)} W[o,i,v,l] * x[b,i,l] * probs[b,o,i]
__global__ void caps_update_kernel(const float* __restrict__ w,
                                   const float* __restrict__ x,
                                   const float* __restrict__ probs,
                                   float* __restrict__ outw) {
  caps_gemm_body<true>(w, x, probs, outw, 1.0f);
}

// ---------------------------------------------------------------------------
// nhat[o,v,b] = out[o,v,b] / max(||out[o,:,b]||, 1e-12)
// ---------------------------------------------------------------------------
__global__ void caps_nhat_kernel(const float* __restrict__ outw,
                                 float* __restrict__ nhat) {
  const int row = blockIdx.x * blockDim.x + threadIdx.x;  // 4096 = O*B
  const int o = row >> 6, b = row & 63;
  const float* p = outw + (size_t)o * NV * BATCH + b;
  float v[NV], ss = 0.0f;
  #pragma unroll
  for (int t = 0; t < NV; ++t) { v[t] = p[(size_t)t * BATCH]; ss += v[t] * v[t]; }
  const float inv = 1.0f / fmaxf(sqrtf(ss), 1e-12f);
  float* q = nhat + (size_t)o * NV * BATCH + b;
  #pragma unroll
  for (int t = 0; t < NV; ++t) q[(size_t)t * BATCH] = v[t] * inv;
}

// ---------------------------------------------------------------------------
// logits[b,o,i] = sum_v priors[b,o,i,v] * nhat[b,o,v]; softmax over o.
// One WG per block of IB=8 in-capsules: 16 waves cover (64 o x 8 i x 4 btile)
// prior tiles (2 WMMAs each), v-reduction in VALU + __shfl_xor(16) (wave32),
// logits in a 128 KB LDS slab (CDNA5: 320 KB/WGP), softmax by 512 threads.
// ---------------------------------------------------------------------------
__global__ void caps_logits_softmax_kernel(const float* __restrict__ w,
                                           const float* __restrict__ x,
                                           const float* __restrict__ nhat,
                                           float* __restrict__ probs) {
  extern __shared__ float slog[];   // [NO][IB][BATCH] = 32768 f32 = 128 KB
  const int tid  = threadIdx.x;     // 512 threads = 16 waves
  const int wv   = tid >> 5;
  const int lane = tid & 31;
  const int half = lane >> 4;
  const int lr   = lane & 15;
  const int ib0  = blockIdx.x * IB;

  for (int t = wv; t < NO * IB * 4; t += 16) {   // uniform trip count: EXEC all-1s
    const int o  = t >> 5;
    const int r2 = t & 31;
    const int ii = r2 >> 2;
    const int bt = r2 & 3;
    const int i  = ib0 + ii;
    const int bb = bt * 16 + lr;
    const float* __restrict__ wrow = w + ((size_t)o * NI + i) * NV * NL; // [v][l]

    v8f c = {};
    v2f a0 = *(const v2f*)(wrow + (size_t)lr * NL + 2 * half);
    v2f b0 = *(const v2f*)(x + ((size_t)bb * NI + i) * NL + 2 * half);
    c = wmma_f32_k4(a0, b0, c);
    v2f a1 = *(const v2f*)(wrow + (size_t)lr * NL + 4 + 2 * half);
    v2f b1 = *(const v2f*)(x + ((size_t)bb * NI + i) * NL + 4 + 2 * half);
    c = wmma_f32_k4(a1, b1, c);

    // weight tile by nhat[o,v,b] and reduce over v (8 regs + cross-half add)
    const float* __restrict__ nh = nhat + (size_t)o * NV * BATCH + bb;
    float acc = 0.0f;
    #pragma unroll
    for (int r = 0; r < 8; ++r) acc += c[r] * nh[(size_t)(r + 8 * half) * BATCH];
    acc += __shfl_xor(acc, 16, 32);          // combine v[0..7] and v[8..15]
    if (half == 0) slog[(o * IB + ii) * BATCH + bb] = acc;
  }
  __syncthreads();

  // softmax over O for each (i, b); exactly one thread per (ii, b)
  const int ii = tid >> 6;
  const int b  = tid & 63;
  const int i  = ib0 + ii;
  float mx = -3.402823466e+38f;
  for (int o = 0; o < NO; ++o)
    mx = fmaxf(mx, slog[(o * IB + ii) * BATCH + b]);
  float s = 0.0f;
  for (int o = 0; o < NO; ++o) {
    const float e = __expf(slog[(o * IB + ii) * BATCH + b] - mx);
    slog[(o * IB + ii) * BATCH + b] = e;
    s += e;
  }
  const float inv = 1.0f / s;
  for (int o = 0; o < NO; ++o)
    probs[((size_t)o * NI + i) * BATCH + b] =
        slog[(o * IB + ii) * BATCH + b] * inv;
}

// ---------------------------------------------------------------------------
// y[b,o,v] = squash(out)[o,v,b] + bias[o,v];  squash scale = n/(1+n^2)
// ---------------------------------------------------------------------------
__global__ void caps_final_kernel(const float* __restrict__ outw,
                                  const float* __restrict__ bias,
                                  float* __restrict__ y) {
  const int row = blockIdx.x * blockDim.x + threadIdx.x;  // 4096 = O*B
  const int o = row >> 6, b = row & 63;
  const float* p = outw + (size_t)o * NV * BATCH + b;
  float v[NV], ss = 0.0f;
  #pragma unroll
  for (int t = 0; t < NV; ++t) { v[t] = p[(size_t)t * BATCH]; ss += v[t] * v[t]; }
  const float n  = sqrtf(ss);
  const float sc = n / (1.0f + ss);   // ||x|| / (1 + ||x||^2)
  #pragma unroll
  for (int t = 0; t < NV; ++t)
    y[((size_t)b * NO + o) * NV + t] = v[t] * sc + bias[o * NV + t];
}

// ---------------------------------------------------------------------------
extern "C" void kernel_launch(void* const* d_in, const int* in_sizes, int n_in,
                              void* d_out, int out_size, void* d_ws, size_t ws_size,
                              hipStream_t stream) {
  const float* x    = (const float*)d_in[0];   // [B, I, L]
  const float* w    = (const float*)d_in[1];   // [O, I, V, L]
  const float* bias = (const float*)d_in[2];   // [O, V]
  float* y = (float*)d_out;                    // [B, O, V]

  float* probs = (float*)d_ws;                         // [O][I][B]   8 MB
  float* outw  = probs + (size_t)NO * NI * BATCH;      // [O][V][B] 256 KB
  float* nhat  = outw  + (size_t)NO * NV * BATCH;      // [O][V][B] 256 KB
  const size_t need =
      ((size_t)NO * NI * BATCH + 2 * (size_t)NO * NV * BATCH) * sizeof(float);
  if (ws_size < need) return;

  caps_init_kernel<<<NO, 128, 0, stream>>>(w, x, outw);
  for (int it = 0; it < 3; ++it) {
    caps_nhat_kernel<<<16, 256, 0, stream>>>(outw, nhat);
    caps_logits_softmax_kernel<<<64, 512,
        (size_t)NO * IB * BATCH * sizeof(float), stream>>>(w, x, nhat, probs);
    caps_update_kernel<<<NO, 128, 0, stream>>>(w, x, probs, outw);
  }
  caps_final_kernel<<<16, 256, 0, stream>>>(outw, bias, y);
}